// NewModel_47029891891468
// MI455X (gfx1250) — compile-verified
//
#include <hip/hip_runtime.h>
#include <hip/hip_bf16.h>

typedef __attribute__((ext_vector_type(2))) float v2f;
typedef __attribute__((ext_vector_type(8))) float v8f;

// ---------------------------------------------------------------------------
// Projection: Y[N,64] = X[N,16] @ W[16,64] + bias[64]
// One wave computes one 16x16 output tile via 4 chained V_WMMA_F32_16X16X4_F32.
// Block = 128 threads (4 waves) covers 16 rows x 64 cols. Grid = ceil(N/16).
// A-row loads are clamped (not predicated): an out-of-range A row only feeds
// the matching out-of-range D row, which is never stored, so clamping is safe
// and keeps the WMMA loop branch-free with unconditional b64 loads.
// ---------------------------------------------------------------------------
__global__ void proj16x64_wmma(const float* __restrict__ X, int N,
                               const float* __restrict__ W,     // [16,64] row-major
                               const float* __restrict__ bias,  // [64]
                               float* __restrict__ Y)           // [N,64]
{
    const int wave = threadIdx.x >> 5;       // 0..3 -> 16-col tile of H
    const int lane = threadIdx.x & 31;
    const int r0   = blockIdx.x * 16;
    const int n0   = wave * 16;
    const int rc   = lane & 15;              // row (A) / col (B,C/D)
    const int khl  = (lane >> 4) * 2;        // 0 for lanes 0-15, 2 for 16-31

    const int gr = min(r0 + rc, N - 1);      // clamped A row (safe, see above)
    const float* __restrict__ Xrow = X + (size_t)gr * 16;
    const float* __restrict__ Wc   = W + n0 + rc;

    v8f c = {};
#pragma unroll
    for (int k0 = 0; k0 < 16; k0 += 4) {
        const int ka = k0 + khl;
        v2f a = *(const v2f*)(Xrow + ka);    // contiguous pair -> global_load_b64
        v2f b;
        b.x = Wc[(size_t)ka * 64];
        b.y = Wc[(size_t)(ka + 1) * 64];
        c = __builtin_amdgcn_wmma_f32_16x16x4_f32(
                /*neg_a=*/false, a, /*neg_b=*/false, b,
                /*c_mod=*/(short)0, c, /*reuse_a=*/false, /*reuse_b=*/false);
    }

    const int n     = n0 + rc;
    const int mbase = (lane >> 4) * 8;       // lanes 16-31 hold M = vgpr+8
    const float bv  = bias[n];
    float* __restrict__ Yp = Y + (size_t)(r0 + mbase) * 64 + n;

    if (r0 + 16 <= N) {                      // block-uniform fast path
#pragma unroll
        for (int m = 0; m < 8; ++m) Yp[(size_t)m * 64] = c[m] + bv;
    } else {
#pragma unroll
        for (int m = 0; m < 8; ++m)
            if (r0 + mbase + m < N) Yp[(size_t)m * 64] = c[m] + bv;
    }
}

// ---------------------------------------------------------------------------
// Edge kernel: one wave per edge, each lane handles 2 of 64 channels.
//   e   = ea * We + be
//   eta = sigmoid(k[dst] + q[src] + 2*e)
//   H[dst] += eta * (v[src] + e)      (atomic f32)
// ---------------------------------------------------------------------------
__global__ void edge_msg_kernel(const int* __restrict__ src, const int* __restrict__ dst,
                                const float* __restrict__ ea,
                                const float* __restrict__ K, const float* __restrict__ Q,
                                const float* __restrict__ V,
                                const float* __restrict__ We, const float* __restrict__ be,
                                float* __restrict__ H, int E)
{
    const int edge = (int)((blockIdx.x * (unsigned)blockDim.x + threadIdx.x) >> 5);
    const int lane = threadIdx.x & 31;
    if (edge >= E) return;

    const int   s = src[edge];
    const int   d = dst[edge];
    const float a = ea[edge];
    const int   h = lane * 2;

    const float2 we = *(const float2*)(We + h);
    const float2 bb = *(const float2*)(be + h);
    const float ex = a * we.x + bb.x;
    const float ey = a * we.y + bb.y;

    const float2 kv = *(const float2*)(K + (size_t)d * 64 + h);
    const float2 qv = *(const float2*)(Q + (size_t)s * 64 + h);
    const float2 vv = *(const float2*)(V + (size_t)s * 64 + h);

    const float gx = kv.x + qv.x + 2.0f * ex;
    const float gy = kv.y + qv.y + 2.0f * ey;
    const float sx = 1.0f / (1.0f + expf(-gx));
    const float sy = 1.0f / (1.0f + expf(-gy));
    const float mx = sx * (vv.x + ex);
    const float my = sy * (vv.y + ey);

    atomicAdd(H + (size_t)d * 64 + h,     mx);
    atomicAdd(H + (size_t)d * 64 + h + 1, my);
}

// ---------------------------------------------------------------------------
// Pool: relu(H[node]) atomically accumulated into per-graph sums + counts.
// One wave per node, 2 channels per lane.
// ---------------------------------------------------------------------------
__global__ void pool_kernel(const float* __restrict__ Hn, const int* __restrict__ batch,
                            int N, float* __restrict__ sums /*[G,64]*/,
                            float* __restrict__ cnt /*[G]*/)
{
    const int node = (int)((blockIdx.x * (unsigned)blockDim.x + threadIdx.x) >> 5);
    const int lane = threadIdx.x & 31;
    if (node >= N) return;
    const int g = batch[node];
    const int h = lane * 2;
    const float2 x = *(const float2*)(Hn + (size_t)node * 64 + h);
    const float rx = x.x > 0.0f ? x.x : 0.0f;
    const float ry = x.y > 0.0f ? x.y : 0.0f;
    atomicAdd(sums + (size_t)g * 64 + h,     rx);
    atomicAdd(sums + (size_t)g * 64 + h + 1, ry);
    if (lane == 0) atomicAdd(cnt + g, 1.0f);
}

// ---------------------------------------------------------------------------
// MLP head: one block (64 threads) per graph.
// pooled = concat(sum_c/cnt_c, sum_b/cnt_b)  [128]
// h1 = relu(pooled@W1+b1); h2 = relu(h1@W2+b2); h3 = relu(h2@W3+b3)
// out = h3@Wout + bout
// ---------------------------------------------------------------------------
__global__ void mlp_kernel(const float* __restrict__ sum_c, const float* __restrict__ cnt_c,
                           const float* __restrict__ sum_b, const float* __restrict__ cnt_b,
                           const float* __restrict__ W1, const float* __restrict__ b1,
                           const float* __restrict__ W2, const float* __restrict__ b2,
                           const float* __restrict__ W3, const float* __restrict__ b3,
                           const float* __restrict__ Wout, const float* __restrict__ bout,
                           float* __restrict__ out)
{
    __shared__ float p[128];
    __shared__ float hbuf[64];
    __shared__ float red[64];
    const int g = blockIdx.x;
    const int t = threadIdx.x;  // 0..63

    const float cc = fmaxf(cnt_c[g], 1.0f);
    const float cb = fmaxf(cnt_b[g], 1.0f);
    p[t]      = sum_c[(size_t)g * 64 + t] / cc;
    p[t + 64] = sum_b[(size_t)g * 64 + t] / cb;
    __syncthreads();

    float acc = b1[t];
    for (int j = 0; j < 128; ++j) acc += p[j] * W1[(size_t)j * 64 + t];
    acc = fmaxf(acc, 0.0f);
    hbuf[t] = acc;
    __syncthreads();

    float acc2 = b2[t];
    for (int j = 0; j < 64; ++j) acc2 += hbuf[j] * W2[(size_t)j * 64 + t];
    acc2 = fmaxf(acc2, 0.0f);
    __syncthreads();
    hbuf[t] = acc2;
    __syncthreads();

    float acc3 = b3[t];
    for (int j = 0; j < 64; ++j) acc3 += hbuf[j] * W3[(size_t)j * 64 + t];
    acc3 = fmaxf(acc3, 0.0f);

    red[t] = acc3 * Wout[t];
    __syncthreads();
    for (int s = 32; s > 0; s >>= 1) {
        if (t < s) red[t] += red[t + s];
        __syncthreads();
    }
    if (t == 0) out[g] = red[0] + bout[0];
}

// ---------------------------------------------------------------------------
extern "C" void kernel_launch(void* const* d_in, const int* in_sizes, int n_in,
                              void* d_out, int out_size, void* d_ws, size_t ws_size,
                              hipStream_t stream) {
    // Inputs in setup_inputs() order
    const float* x_x   = (const float*)d_in[0];
    const float* x_c   = (const float*)d_in[1];
    const float* x_b   = (const float*)d_in[2];
    const float* ea_ac = (const float*)d_in[3];
    const float* ea_cb = (const float*)d_in[4];
    const float* Wk_ac = (const float*)d_in[5];
    const float* Wq_ac = (const float*)d_in[6];
    const float* Wv_ac = (const float*)d_in[7];
    const float* Ws_ac = (const float*)d_in[8];
    const float* We_ac = (const float*)d_in[9];
    const float* bk_ac = (const float*)d_in[10];
    const float* bq_ac = (const float*)d_in[11];
    const float* bv_ac = (const float*)d_in[12];
    const float* be_ac = (const float*)d_in[13];
    const float* bc_ac = (const float*)d_in[14];
    const float* Wk_cb = (const float*)d_in[15];
    const float* Wq_cb = (const float*)d_in[16];
    const float* Wv_cb = (const float*)d_in[17];
    const float* Ws_cb = (const float*)d_in[18];
    const float* We_cb = (const float*)d_in[19];
    const float* bk_cb = (const float*)d_in[20];
    const float* bq_cb = (const float*)d_in[21];
    const float* bv_cb = (const float*)d_in[22];
    const float* be_cb = (const float*)d_in[23];
    const float* bc_cb = (const float*)d_in[24];
    const float* W1    = (const float*)d_in[25];
    const float* b1    = (const float*)d_in[26];
    const float* W2    = (const float*)d_in[27];
    const float* b2    = (const float*)d_in[28];
    const float* W3    = (const float*)d_in[29];
    const float* b3    = (const float*)d_in[30];
    const float* Wout  = (const float*)d_in[31];
    const float* bout  = (const float*)d_in[32];
    const int* src_ac  = (const int*)d_in[33];
    const int* dst_ac  = (const int*)d_in[34];
    const int* src_cb  = (const int*)d_in[35];
    const int* dst_cb  = (const int*)d_in[36];
    const int* batch_c = (const int*)d_in[37];
    const int* batch_b = (const int*)d_in[38];

    const int Nx = in_sizes[0] / 16;
    const int Nc = in_sizes[1] / 16;
    const int Nb = in_sizes[2] / 16;
    const int E1 = in_sizes[3];
    const int E2 = in_sizes[4];
    const int G  = out_size;   // 128

    // Workspace layout (floats)
    float* ws = (float*)d_ws;
    size_t off = 0;
    float* Qx = ws + off; off += (size_t)Nx * 64;
    float* Vx = ws + off; off += (size_t)Nx * 64;
    float* Kc = ws + off; off += (size_t)Nc * 64;
    float* Qc = ws + off; off += (size_t)Nc * 64;
    float* Vc = ws + off; off += (size_t)Nc * 64;
    float* Kb = ws + off; off += (size_t)Nb * 64;
    float* Hc = ws + off; off += (size_t)Nc * 64;   // skip-init + agg for 'c'
    float* Hb = ws + off; off += (size_t)Nb * 64;   // skip-init + agg for 'b'
    float* sum_c = ws + off; off += (size_t)G * 64;
    float* sum_b = ws + off; off += (size_t)G * 64;
    float* cnt_c = ws + off; off += (size_t)G;
    float* cnt_b = ws + off; off += (size_t)G;

    // Zero pooled sums + counts (contiguous region)
    hipMemsetAsync(sum_c, 0, ((size_t)G * 128 + 2 * (size_t)G) * sizeof(float), stream);

    // Projections (WMMA f32 path)
    const int TB = 128;  // 4 waves
    dim3 gx((Nx + 15) / 16), gc((Nc + 15) / 16), gb((Nb + 15) / 16);
    proj16x64_wmma<<<gx, TB, 0, stream>>>(x_x, Nx, Wq_ac, bq_ac, Qx);
    proj16x64_wmma<<<gx, TB, 0, stream>>>(x_x, Nx, Wv_ac, bv_ac, Vx);
    proj16x64_wmma<<<gc, TB, 0, stream>>>(x_c, Nc, Wk_ac, bk_ac, Kc);
    proj16x64_wmma<<<gc, TB, 0, stream>>>(x_c, Nc, Ws_ac, bc_ac, Hc);  // skip + bconv
    proj16x64_wmma<<<gc, TB, 0, stream>>>(x_c, Nc, Wq_cb, bq_cb, Qc);
    proj16x64_wmma<<<gc, TB, 0, stream>>>(x_c, Nc, Wv_cb, bv_cb, Vc);
    proj16x64_wmma<<<gb, TB, 0, stream>>>(x_b, Nb, Wk_cb, bk_cb, Kb);
    proj16x64_wmma<<<gb, TB, 0, stream>>>(x_b, Nb, Ws_cb, bc_cb, Hb);  // skip + bconv

    // Edge message passing (dominant, memory-bound)
    dim3 ge1((E1 + 7) / 8), ge2((E2 + 7) / 8);
    edge_msg_kernel<<<ge1, 256, 0, stream>>>(src_ac, dst_ac, ea_ac, Kc, Qx, Vx,
                                             We_ac, be_ac, Hc, E1);
    edge_msg_kernel<<<ge2, 256, 0, stream>>>(src_cb, dst_cb, ea_cb, Kb, Qc, Vc,
                                             We_cb, be_cb, Hb, E2);

    // ReLU + segment-sum pooling
    dim3 gp1((Nc + 7) / 8), gp2((Nb + 7) / 8);
    pool_kernel<<<gp1, 256, 0, stream>>>(Hc, batch_c, Nc, sum_c, cnt_c);
    pool_kernel<<<gp2, 256, 0, stream>>>(Hb, batch_b, Nb, sum_b, cnt_b);

    // MLP head
    mlp_kernel<<<G, 64, 0, stream>>>(sum_c, cnt_c, sum_b, cnt_b,
                                     W1, b1, W2, b2, W3, b3, Wout, bout,
                                     (float*)d_out);
}